// DCNv2BNAct_22058952032384
// MI455X (gfx1250) — compile-verified
//
#include <hip/hip_runtime.h>
#include <math.h>

typedef __attribute__((ext_vector_type(2))) float v2f;
typedef __attribute__((ext_vector_type(8))) float v8f;

// Problem constants (match reference)
constexpr int BATCH = 2;
constexpr int C1c   = 256;
constexpr int C2c   = 256;
constexpr int Hc    = 64;
constexpr int Wc    = 64;
constexpr int K2c   = 9;
constexpr int KDIM  = C1c * K2c;        // 2304
constexpr int Nc    = BATCH * Hc * Wc;  // 8192
constexpr int KB    = 32;               // K-slab for LDS staging in main GEMM

__device__ __forceinline__ float sigmoidf_(float v) {
    return 1.0f / (1.0f + __expf(-v));
}

// ---------------------------------------------------------------------------
// Prep 1: w_offT[r][o32] = w_off[o][c][t], K-order r = t*256 + c, o padded to 32
// with zeros (removes row masking from the WMMA loop entirely).
// ---------------------------------------------------------------------------
__global__ void __launch_bounds__(256) prep_w_offT(
    const float* __restrict__ w_off, float* __restrict__ w_offT) {
    int id = blockIdx.x * 256 + threadIdx.x;   // < 2304*32
    int r  = id >> 5, o = id & 31;
    int c  = r & 255, t = r >> 8;
    w_offT[id] = (o < 27) ? w_off[(o * 256 + c) * 9 + t] : 0.0f;
}

// ---------------------------------------------------------------------------
// Prep 2: w_dcnT[r][o] = w_dcn[o][c][k], K-order r = k*256 + c.
// Makes A-fragment loads lane-consecutive (coalesced).
// ---------------------------------------------------------------------------
__global__ void __launch_bounds__(256) prep_w_dcnT(
    const float* __restrict__ w_dcn, float* __restrict__ w_dcnT) {
    int id = blockIdx.x * 256 + threadIdx.x;   // < 2304*256
    int r  = id >> 8, o = id & 255;
    int c  = r & 255, k = r >> 8;
    w_dcnT[id] = w_dcn[(o * 256 + c) * 9 + k];
}

// ---------------------------------------------------------------------------
// Kernel 1: offset conv as WMMA f32 GEMM, tap-major K order.
// A = w_offT (padded/transposed), B = im2col(x) gathered with per-tap hoisted
// geometry.  Wave: 2 M-tiles x 4 N-tiles (32 x 64).  Block = 4 waves.
// ---------------------------------------------------------------------------
__global__ void __launch_bounds__(128) dcn_offset_conv_wmma(
    const float* __restrict__ x, const float* __restrict__ w_offT,
    const float* __restrict__ b_off, float* __restrict__ om) {
    const int lane   = threadIdx.x & 31;
    const int wave   = threadIdx.x >> 5;
    const int n_base = blockIdx.x * 256 + wave * 64;
    const int l15    = lane & 15;
    const int khalf  = (lane >> 4) << 1;  // 0 or 2

    v8f acc[2][4];
#pragma unroll
    for (int mt = 0; mt < 2; ++mt)
#pragma unroll
        for (int nt = 0; nt < 4; ++nt) acc[mt][nt] = v8f{};

    // Per-N-tile pixel geometry (fixed for whole kernel)
    int hv[4], wv[4];
    const float* xbase[4];
#pragma unroll
    for (int nt = 0; nt < 4; ++nt) {
        int col = n_base + nt * 16 + l15;
        int b = col >> 12, hw = col & 4095;
        hv[nt] = hw >> 6;
        wv[nt] = hw & 63;
        xbase[nt] = x + (b << 20);
    }

    for (int t = 0; t < 9; ++t) {
        const int ty = t / 3, tx = t - ty * 3;
        const float* xt[4];
        float okv[4];
#pragma unroll
        for (int nt = 0; nt < 4; ++nt) {
            int y  = hv[nt] + ty - 1;
            int xx = wv[nt] + tx - 1;
            okv[nt] = (((unsigned)y < 64u) && ((unsigned)xx < 64u)) ? 1.0f : 0.0f;
            int yc = min(max(y, 0), 63);
            int xc = min(max(xx, 0), 63);
            xt[nt] = xbase[nt] + (yc << 6) + xc;
        }
        const int kbase = t << 8;
        for (int c0 = 0; c0 < 256; c0 += 4) {
            const int kk = kbase + c0 + khalf;
            v2f a[2];
#pragma unroll
            for (int mt = 0; mt < 2; ++mt) {
                int rowi = mt * 16 + l15;
                a[mt].x = w_offT[kk * 32 + rowi];
                a[mt].y = w_offT[(kk + 1) * 32 + rowi];
            }
            const int coff = (c0 + khalf) << 12;  // channel plane offset
#pragma unroll
            for (int nt = 0; nt < 4; ++nt) {
                v2f bfr;
                bfr.x = xt[nt][coff] * okv[nt];
                bfr.y = xt[nt][coff + 4096] * okv[nt];
#pragma unroll
                for (int mt = 0; mt < 2; ++mt)
                    acc[mt][nt] = __builtin_amdgcn_wmma_f32_16x16x4_f32(
                        false, a[mt], false, bfr, (short)0, acc[mt][nt], false, false);
            }
        }
    }

#pragma unroll
    for (int mt = 0; mt < 2; ++mt) {
#pragma unroll
        for (int v = 0; v < 8; ++v) {
            int o = mt * 16 + v + ((lane >> 4) << 3);
            if (o < 27) {
                float bias = b_off[o];
#pragma unroll
                for (int nt = 0; nt < 4; ++nt) {
                    int col = n_base + nt * 16 + l15;
                    om[o * Nc + col] = acc[mt][nt][v] + bias;
                }
            }
        }
    }
}

// ---------------------------------------------------------------------------
// Kernel 2: deformable bilinear sampling -> vals[(k*256+c)*8192 + n]
// (tap-major K order matching w_dcnT).
// ---------------------------------------------------------------------------
__global__ void __launch_bounds__(256) dcn_sample(
    const float* __restrict__ x, const float* __restrict__ om,
    float* __restrict__ vals) {
    const int id = blockIdx.x * 256 + threadIdx.x;  // 0 .. 9*8192-1
    const int k  = id >> 13;
    const int n  = id & (Nc - 1);
    const int ty = k / 3, tx = k - ty * 3;
    const int b = n >> 12, hw = n & 4095, h = hw >> 6, w = hw & 63;

    float dy = om[(2 * k) * Nc + n];
    float dx = om[(2 * k + 1) * Nc + n];
    float m  = sigmoidf_(om[(18 + k) * Nc + n]);

    float py  = (float)(h - 1 + ty) + dy;
    float px  = (float)(w - 1 + tx) + dx;
    float y0f = floorf(py), x0f = floorf(px);
    float ly  = py - y0f,   lx  = px - x0f;
    int y0 = (int)y0f, x0 = (int)x0f;

    float vy0 = ((unsigned)y0 < 64u) ? 1.0f : 0.0f;
    float vy1 = ((unsigned)(y0 + 1) < 64u) ? 1.0f : 0.0f;
    float vx0 = ((unsigned)x0 < 64u) ? 1.0f : 0.0f;
    float vx1 = ((unsigned)(x0 + 1) < 64u) ? 1.0f : 0.0f;

    float w00 = (1.0f - ly) * (1.0f - lx) * vy0 * vx0 * m;
    float w01 = (1.0f - ly) * lx          * vy0 * vx1 * m;
    float w10 = ly          * (1.0f - lx) * vy1 * vx0 * m;
    float w11 = ly          * lx          * vy1 * vx1 * m;

    int y0c = min(max(y0, 0), 63),     y1c = min(max(y0 + 1, 0), 63);
    int x0c = min(max(x0, 0), 63),     x1c = min(max(x0 + 1, 0), 63);
    int i00 = (y0c << 6) + x0c, i01 = (y0c << 6) + x1c;
    int i10 = (y1c << 6) + x0c, i11 = (y1c << 6) + x1c;

    const float* xb = x + (b << 20);
    float* outp = vals + (k << 8) * Nc + n;   // row r = k*256 + c
#pragma unroll 4
    for (int c = 0; c < C1c; ++c) {
        const float* xc = xb + (c << 12);
        float v = w00 * xc[i00] + w01 * xc[i01] + w10 * xc[i10] + w11 * xc[i11];
        outp[c * Nc] = v;
    }
}

// ---------------------------------------------------------------------------
// Kernel 3: main GEMM out[o][n] = sum_r w_dcnT[r][o] * vals[r][n].
// Double-buffered ASYNC global->LDS staging (global_load_async_to_lds_b128,
// ASYNCcnt) overlapped with WMMA compute; fused BN + SiLU epilogue.
// Block: 8 waves = 128M x 256N.  Grid: (8192/256, 256/128) = (32, 2).
// LDS: 2*(16KB A + 32KB B) = 96KB of the WGP's 320KB.
// ---------------------------------------------------------------------------
__global__ void __launch_bounds__(256) dcn_gemm_bn_silu(
    const float* __restrict__ w_dcnT, const float* __restrict__ vals,
    const float* __restrict__ gamma, const float* __restrict__ beta,
    const float* __restrict__ rmean, const float* __restrict__ rvar,
    float* __restrict__ out) {
    __shared__ __align__(16) float As[2][KB * 128];
    __shared__ __align__(16) float Bs[2][KB * 256];

    const int tid   = threadIdx.x;
    const int lane  = tid & 31;
    const int wave  = tid >> 5;          // 0..7
    const int wm    = wave >> 2, wn = wave & 3;
    const int l15   = lane & 15;
    const int khalf = (lane >> 4) << 1;
    const int m_blk = blockIdx.y * 128;
    const int n_blk = blockIdx.x * 256;

    // LDS byte offsets of the two buffers (low 32 bits of a shared-aperture
    // flat address are the LDS byte offset).
    unsigned aBase[2], bBase[2];
    aBase[0] = (unsigned)(unsigned long long)(uintptr_t)&As[0][0];
    aBase[1] = (unsigned)(unsigned long long)(uintptr_t)&As[1][0];
    bBase[0] = (unsigned)(unsigned long long)(uintptr_t)&Bs[0][0];
    bBase[1] = (unsigned)(unsigned long long)(uintptr_t)&Bs[1][0];

    // Issue async global->LDS copies for one K-slab into buffer `bufsel`.
    auto stage = [&](int k0, int bufsel) {
        // A slab: KB x 128 floats = 1024 float4 (4 per thread)
#pragma unroll
        for (int j = 0; j < 4; ++j) {
            int q = tid + j * 256;
            int row = q >> 5, c4 = q & 31;
            unsigned lo = aBase[bufsel] + (unsigned)((row * 128 + c4 * 4) * 4);
            const float* g = w_dcnT + (k0 + row) * 256 + m_blk + c4 * 4;
            asm volatile("global_load_async_to_lds_b128 %0, %1, off"
                         :: "v"(lo), "v"(g) : "memory");
        }
        // B slab: KB x 256 floats = 2048 float4 (8 per thread)
#pragma unroll
        for (int j = 0; j < 8; ++j) {
            int q = tid + j * 256;
            int row = q >> 6, c4 = q & 63;
            unsigned lo = bBase[bufsel] + (unsigned)((row * 256 + c4 * 4) * 4);
            const float* g = vals + (k0 + row) * Nc + n_blk + c4 * 4;
            asm volatile("global_load_async_to_lds_b128 %0, %1, off"
                         :: "v"(lo), "v"(g) : "memory");
        }
    };

    v8f acc[4][4];
#pragma unroll
    for (int mt = 0; mt < 4; ++mt)
#pragma unroll
        for (int nt = 0; nt < 4; ++nt) acc[mt][nt] = v8f{};

    constexpr int NSLAB = KDIM / KB;  // 72
    stage(0, 0);

    for (int s = 0; s < NSLAB; ++s) {
        // Publish slab s (our async copies done) and make sure every thread
        // finished reading the buffer we are about to overwrite.
        asm volatile("s_wait_asynccnt 0x0" ::: "memory");
        __syncthreads();
        if (s + 1 < NSLAB) stage((s + 1) * KB, (s + 1) & 1);

        const float* Asb = &As[s & 1][0];
        const float* Bsb = &Bs[s & 1][0];
#pragma unroll
        for (int kl = 0; kl < KB; kl += 4) {
            const int kk = kl + khalf;
            v2f a[4];
#pragma unroll
            for (int mt = 0; mt < 4; ++mt) {
                int rowi = wm * 64 + mt * 16 + l15;
                a[mt].x = Asb[kk * 128 + rowi];
                a[mt].y = Asb[(kk + 1) * 128 + rowi];
            }
            v2f bb[4];
#pragma unroll
            for (int nt = 0; nt < 4; ++nt) {
                int coli = wn * 64 + nt * 16 + l15;
                bb[nt].x = Bsb[kk * 256 + coli];
                bb[nt].y = Bsb[(kk + 1) * 256 + coli];
            }
#pragma unroll
            for (int mt = 0; mt < 4; ++mt)
#pragma unroll
                for (int nt = 0; nt < 4; ++nt)
                    acc[mt][nt] = __builtin_amdgcn_wmma_f32_16x16x4_f32(
                        false, a[mt], false, bb[nt], (short)0, acc[mt][nt], false, false);
        }
    }

    // Epilogue: BN affine + SiLU, store NCHW.
#pragma unroll
    for (int mt = 0; mt < 4; ++mt) {
#pragma unroll
        for (int v = 0; v < 8; ++v) {
            int o = m_blk + wm * 64 + mt * 16 + v + ((lane >> 4) << 3);
            float inv   = gamma[o] / sqrtf(rvar[o] + 1e-5f);
            float shift = beta[o] - rmean[o] * inv;
#pragma unroll
            for (int nt = 0; nt < 4; ++nt) {
                int n = n_blk + wn * 64 + nt * 16 + l15;
                float yv = acc[mt][nt][v] * inv + shift;
                float r  = yv * sigmoidf_(yv);
                int b = n >> 12, hw = n & 4095;
                out[(((b << 8) + o) << 12) + hw] = r;
            }
        }
    }
}

extern "C" void kernel_launch(void* const* d_in, const int* in_sizes, int n_in,
                              void* d_out, int out_size, void* d_ws, size_t ws_size,
                              hipStream_t stream) {
    const float* x      = (const float*)d_in[0];
    const float* w_off  = (const float*)d_in[1];
    const float* b_off  = (const float*)d_in[2];
    const float* w_dcn  = (const float*)d_in[3];
    const float* gamma  = (const float*)d_in[4];
    const float* beta   = (const float*)d_in[5];
    const float* rmean  = (const float*)d_in[6];
    const float* rvar   = (const float*)d_in[7];
    float* out = (float*)d_out;

    // Workspace layout (floats): om[27*8192] | w_offT[2304*32] | w_dcnT[2304*256]
    //                            | vals[2304*8192]  (~79 MB total, L2-resident)
    float* om     = (float*)d_ws;
    float* w_offT = om + 27 * Nc;
    float* w_dcnT = w_offT + KDIM * 32;
    float* vals   = w_dcnT + KDIM * 256;

    prep_w_offT<<<(KDIM * 32) / 256, 256, 0, stream>>>(w_off, w_offT);
    prep_w_dcnT<<<(KDIM * 256) / 256, 256, 0, stream>>>(w_dcn, w_dcnT);
    dcn_offset_conv_wmma<<<Nc / 256, 128, 0, stream>>>(x, w_offT, b_off, om);
    dcn_sample<<<(K2c * Nc) / 256, 256, 0, stream>>>(x, om, vals);
    dim3 g3(Nc / 256, C2c / 128);
    dcn_gemm_bn_silu<<<g3, 256, 0, stream>>>(w_dcnT, vals, gamma, beta, rmean, rvar, out);
}